// ProposalLayer_8014408974406
// MI455X (gfx1250) — compile-verified
//
#include <hip/hip_runtime.h>
#include <stdint.h>

// ---------------------------------------------------------------------------
// ProposalLayer for MI455X (gfx1250, wave32)
//   stage 1: exact radix-select of the 6000th-largest score (4 x 8-bit passes)
//   stage 2: compact candidates (score >= T), bitonic sort 8192 keys in LDS,
//            decode boxes (deltas*std, exp, clip) in score-descending order
//   stage 3: suppression bitmask: one wave32 per 16x32 tile.
//            pairwise (area_i + area_j) via V_WMMA_F32_16X16X4_F32,
//            tile boxes staged in LDS via global_load_async_to_lds_b128.
//   stage 4: per-batch sequential OR-reduce of mask rows (early exit @1000),
//            write 1000x4 proposals (zero padded).
// ---------------------------------------------------------------------------

#define BATCH 4
#define NANCH 261888
#define KSEL  6000
#define PROP  1000
#define CANDN 8192
#define MWU   188          // ceil(6000/32) mask words per row
#define NMS_THR 0.7f

typedef __attribute__((ext_vector_type(2))) float v2f;
typedef __attribute__((ext_vector_type(8))) float v8f;
typedef __attribute__((ext_vector_type(4))) int   v4i;

#if defined(__gfx1250__) && __has_builtin(__builtin_amdgcn_global_load_async_to_lds_b128)
#define HAVE_ASYNC_LDS 1
typedef __attribute__((address_space(1))) v4i gv4i;   // global int4
typedef __attribute__((address_space(3))) v4i lv4i;   // LDS int4
#else
#define HAVE_ASYNC_LDS 0
#endif

#if defined(__gfx1250__) && __has_builtin(__builtin_amdgcn_wmma_f32_16x16x4_f32)
#define HAVE_WMMA 1
#else
#define HAVE_WMMA 0
#endif

__device__ __forceinline__ float clip01(float v) { return fminf(fmaxf(v, 0.f), 1.f); }

// ---------------- stage 0: init workspace state ----------------------------
__global__ void init_kernel(unsigned* state, unsigned* hist) {
  int t = threadIdx.x;
  for (int i = t; i < BATCH * 256; i += blockDim.x) hist[i] = 0u;
  if (t < BATCH) {
    state[t * 4 + 0] = 0u;            // radix prefix / final threshold T
    state[t * 4 + 1] = (unsigned)KSEL;// k remaining within prefix class
    state[t * 4 + 2] = 0u;            // candidate count
    state[t * 4 + 3] = 0u;
  }
}

// ---------------- stage 1a: 256-bin histogram of current byte --------------
__global__ __launch_bounds__(256) void hist_kernel(const float* __restrict__ classes,
                                                   unsigned* __restrict__ hist,
                                                   const unsigned* __restrict__ state,
                                                   int pass) {
  __shared__ unsigned lh[256];
  const int b = blockIdx.y;
  lh[threadIdx.x] = 0u;
  __syncthreads();
  const unsigned prefix = state[b * 4 + 0];
  const unsigned msk = pass ? (0xFFFFFFFFu << (32 - 8 * pass)) : 0u;
  const int shift = 24 - 8 * pass;
  const float2* sc = (const float2*)classes + (size_t)b * NANCH;
  for (int i = blockIdx.x * blockDim.x + threadIdx.x; i < NANCH; i += blockDim.x * gridDim.x) {
    unsigned u = __float_as_uint(sc[i].y);  // foreground score (>=0, bit-order == value-order)
    if ((u & msk) == prefix) atomicAdd(&lh[(u >> shift) & 255u], 1u);
  }
  __syncthreads();
  unsigned c = lh[threadIdx.x];
  if (c) atomicAdd(&hist[b * 256 + threadIdx.x], c);
}

// ---------------- stage 1b: pick byte containing the k-th largest ----------
__global__ void scan_kernel(unsigned* hist, unsigned* state, int pass) {
  const int b = blockIdx.x;
  unsigned* h = hist + b * 256;
  if (threadIdx.x == 0) {
    unsigned kcur = state[b * 4 + 1];
    unsigned cum = 0; int sel = 0;
    for (int v = 255; v >= 0; --v) {
      unsigned c = h[v];
      if (cum + c >= kcur) { sel = v; break; }
      cum += c;
    }
    state[b * 4 + 0] |= ((unsigned)sel) << (24 - 8 * pass);
    state[b * 4 + 1] = kcur - cum;   // rank within equal-valued elements
  }
  __syncthreads();
  h[threadIdx.x] = 0u;               // ready for next pass
}

// ---------------- stage 2a: compact candidates with score >= T -------------
__global__ __launch_bounds__(256) void compact_kernel(const float* __restrict__ classes,
                                                      unsigned* __restrict__ state,
                                                      unsigned long long* __restrict__ cand) {
  const int b = blockIdx.y;
  const unsigned T = state[b * 4 + 0];
  const float2* sc = (const float2*)classes + (size_t)b * NANCH;
  for (int i = blockIdx.x * blockDim.x + threadIdx.x; i < NANCH; i += blockDim.x * gridDim.x) {
    unsigned u = __float_as_uint(sc[i].y);
    if (u >= T) {
      unsigned slot = atomicAdd(&state[b * 4 + 2], 1u);
      if (slot < CANDN)   // key: score desc, then index asc (jax.top_k tie order)
        cand[(size_t)b * CANDN + slot] =
            ((unsigned long long)u << 32) | (unsigned)(~(unsigned)i);
    }
  }
}

// ---------------- stage 2b: bitonic sort (desc) + box decode ---------------
__global__ __launch_bounds__(1024) void sort_decode_kernel(const unsigned long long* __restrict__ cand,
                                                           const float* __restrict__ anchors,
                                                           const float* __restrict__ bboxes,
                                                           const unsigned* __restrict__ state,
                                                           float* __restrict__ dec) {
  extern __shared__ unsigned long long key[];   // 8192 * 8B = 64 KB LDS
  const int b = blockIdx.x;
  const int tid = threadIdx.x;
  unsigned cnt = state[b * 4 + 2];
  if (cnt > CANDN) cnt = CANDN;
  for (int i = tid; i < CANDN; i += blockDim.x)
    key[i] = (i < (int)cnt) ? cand[(size_t)b * CANDN + i] : 0ull;

  for (unsigned sz = 2; sz <= CANDN; sz <<= 1) {
    for (unsigned st = sz >> 1; st > 0; st >>= 1) {
      __syncthreads();
      for (unsigned i = tid; i < CANDN; i += blockDim.x) {
        unsigned j = i ^ st;
        if (j > i) {
          unsigned long long a = key[i], c = key[j];
          bool desc = ((i & sz) == 0);
          if (desc ? (a < c) : (a > c)) { key[i] = c; key[j] = a; }
        }
      }
    }
  }
  __syncthreads();

  for (int p = tid; p < KSEL; p += blockDim.x) {
    unsigned idx = ~((unsigned)key[p]);
    const float* a = anchors + ((size_t)b * NANCH + idx) * 4;
    const float* d = bboxes  + ((size_t)b * NANCH + idx) * 4;
    float dy = d[0] * 0.1f, dx = d[1] * 0.1f, dh = d[2] * 0.2f, dw = d[3] * 0.2f;
    float h = a[2] - a[0], w = a[3] - a[1];
    float cy = a[0] + 0.5f * h + dy * h;
    float cx = a[1] + 0.5f * w + dx * w;
    float h2 = h * expf(dh), w2 = w * expf(dw);
    float* o = dec + ((size_t)b * KSEL + p) * 4;
    o[0] = clip01(cy - 0.5f * h2);
    o[1] = clip01(cx - 0.5f * w2);
    o[2] = clip01(cy + 0.5f * h2);
    o[3] = clip01(cx + 0.5f * w2);
  }
}

// ---------------- stage 3: suppression bitmask, 1 wave per 16x32 tile ------
__global__ __launch_bounds__(32) void mask_kernel(const float* __restrict__ dec,
                                                  unsigned* __restrict__ mask) {
  const int b = blockIdx.z;
  const int rowbase = blockIdx.y << 4;   // 375 * 16 == 6000 exactly
  const int colblk  = blockIdx.x;        // 188 blocks of 32 columns
  const int colbase = colblk << 5;
  const int lane = threadIdx.x;
  const float* boxes = dec + (size_t)b * KSEL * 4;

  __shared__ float srow[16 * 4];
  __shared__ float scol[32 * 4];

  const int cidx = colbase + lane;
  const bool cvalid = cidx < KSEL;
  if (!cvalid) {  // pad cols: zero box -> inter=0 -> never suppresses
    scol[lane * 4 + 0] = 0.f; scol[lane * 4 + 1] = 0.f;
    scol[lane * 4 + 2] = 0.f; scol[lane * 4 + 3] = 0.f;
  }
#if HAVE_ASYNC_LDS
  if (cvalid) {
    __builtin_amdgcn_global_load_async_to_lds_b128(
        (gv4i*)(boxes + (size_t)cidx * 4),
        (lv4i*)(&scol[lane * 4]), 0, 0);
  }
  if (lane < 16) {
    __builtin_amdgcn_global_load_async_to_lds_b128(
        (gv4i*)(boxes + (size_t)(rowbase + lane) * 4),
        (lv4i*)(&srow[lane * 4]), 0, 0);
  }
  asm volatile("s_wait_asynccnt 0" ::: "memory");
#else
  if (cvalid) {
    scol[lane * 4 + 0] = boxes[(size_t)cidx * 4 + 0];
    scol[lane * 4 + 1] = boxes[(size_t)cidx * 4 + 1];
    scol[lane * 4 + 2] = boxes[(size_t)cidx * 4 + 2];
    scol[lane * 4 + 3] = boxes[(size_t)cidx * 4 + 3];
  }
  if (lane < 16) {
    srow[lane * 4 + 0] = boxes[(size_t)(rowbase + lane) * 4 + 0];
    srow[lane * 4 + 1] = boxes[(size_t)(rowbase + lane) * 4 + 1];
    srow[lane * 4 + 2] = boxes[(size_t)(rowbase + lane) * 4 + 2];
    srow[lane * 4 + 3] = boxes[(size_t)(rowbase + lane) * 4 + 3];
  }
#endif
  __syncthreads();

  const int l15 = lane & 15;
  const bool lo = lane < 16;
  // this lane's fixed column boxes (tile0: n=l15, tile1: n=l15+16)
  const float c0y1 = scol[l15 * 4 + 0], c0x1 = scol[l15 * 4 + 1];
  const float c0y2 = scol[l15 * 4 + 2], c0x2 = scol[l15 * 4 + 3];
  const float c1y1 = scol[(l15 + 16) * 4 + 0], c1x1 = scol[(l15 + 16) * 4 + 1];
  const float c1y2 = scol[(l15 + 16) * 4 + 2], c1x2 = scol[(l15 + 16) * 4 + 3];
  const float cA0 = (c0y2 - c0y1) * (c0x2 - c0x1);
  const float cA1 = (c1y2 - c1y1) * (c1x2 - c1x1);
  const float rAy1 = srow[l15 * 4 + 0], rAx1 = srow[l15 * 4 + 1];
  const float rAy2 = srow[l15 * 4 + 2], rAx2 = srow[l15 * 4 + 3];
  const float rA = (rAy2 - rAy1) * (rAx2 - rAx1);

#if HAVE_WMMA
  // S[m][n] = area_m * 1 + 1 * area_n via rank-2 V_WMMA_F32_16X16X4_F32.
  // A (16x4): lanes 0-15 -> {K0=area_row, K1=1}; lanes 16-31 -> {K2=0, K3=0}
  // B (4x16): lanes 0-15 -> {K0=1, K1=area_col}; lanes 16-31 -> 0
  v2f A;  A.x  = lo ? rA  : 0.f;  A.y  = lo ? 1.f : 0.f;
  v2f B0; B0.x = lo ? 1.f : 0.f;  B0.y = lo ? cA0 : 0.f;
  v2f B1; B1.x = lo ? 1.f : 0.f;  B1.y = lo ? cA1 : 0.f;
  v8f Z = {0.f, 0.f, 0.f, 0.f, 0.f, 0.f, 0.f, 0.f};
  v8f S0 = __builtin_amdgcn_wmma_f32_16x16x4_f32(false, A, false, B0, (short)0, Z, false, false);
  v8f S1 = __builtin_amdgcn_wmma_f32_16x16x4_f32(false, A, false, B1, (short)0, Z, false, false);
#endif

  const int mh = (lane >> 4) << 3;   // C/D layout: lane row = r (lo) or r+8 (hi)
  unsigned myword = 0u;
#pragma unroll
  for (int r = 0; r < 8; ++r) {
    const int m = r + mh;
    const float ry1 = srow[m * 4 + 0], rx1 = srow[m * 4 + 1];
    const float ry2 = srow[m * 4 + 2], rx2 = srow[m * 4 + 3];
#if HAVE_WMMA
    const float sum0 = S0[r];
    const float sum1 = S1[r];
#else
    const float ra = (ry2 - ry1) * (rx2 - rx1);
    const float sum0 = ra + cA0;
    const float sum1 = ra + cA1;
#endif
    float ih0 = fmaxf(fminf(ry2, c0y2) - fmaxf(ry1, c0y1), 0.f);
    float iw0 = fmaxf(fminf(rx2, c0x2) - fmaxf(rx1, c0x1), 0.f);
    float in0 = ih0 * iw0;
    float ih1 = fmaxf(fminf(ry2, c1y2) - fmaxf(ry1, c1y1), 0.f);
    float iw1 = fmaxf(fminf(rx2, c1x2) - fmaxf(rx1, c1x1), 0.f);
    float in1 = ih1 * iw1;
    // iou > thr  <=>  inter > thr*(sumA - inter + 1e-9)
    int sup0 = in0 > NMS_THR * (sum0 - in0 + 1e-9f);
    int sup1 = in1 > NMS_THR * (sum1 - in1 + 1e-9f);
    unsigned long long b0 = __ballot(sup0);
    unsigned long long b1 = __ballot(sup1);
    unsigned wlo = (unsigned)((b0 & 0xFFFFull) | ((b1 & 0xFFFFull) << 16));           // row r
    unsigned whi = (unsigned)(((b0 >> 16) & 0xFFFFull) | (((b1 >> 16) & 0xFFFFull) << 16)); // row r+8
    myword = (lane == r)     ? wlo : myword;
    myword = (lane == r + 8) ? whi : myword;
  }
  if (lane < 16)
    mask[((size_t)b * KSEL + rowbase + lane) * MWU + colblk] = myword;
}

// ---------------- stage 4: sequential greedy reduce + output ---------------
__global__ __launch_bounds__(256) void nms_out_kernel(const float* __restrict__ dec,
                                                      const unsigned* __restrict__ mask,
                                                      float* __restrict__ out) {
  const int b = blockIdx.x;
  const int tid = threadIdx.x;
  __shared__ unsigned removed[MWU];
  __shared__ int keep[PROP];
  __shared__ int kc;
  for (int t = tid; t < MWU; t += blockDim.x) removed[t] = 0u;
  if (tid == 0) kc = 0;
  __syncthreads();

  for (int i = 0; i < KSEL; ++i) {
    bool rem = (removed[i >> 5] >> (i & 31)) & 1u;   // uniform across block
    if (!rem) {
      if (tid == 0) keep[kc] = i;
      const unsigned* row = mask + ((size_t)b * KSEL + i) * MWU;
      for (int t = tid; t < MWU; t += blockDim.x) removed[t] |= row[t];
      __syncthreads();
      if (tid == 0) kc++;
      __syncthreads();
      if (kc >= PROP) break;                          // only first 1000 emitted
    }
  }
  __syncthreads();

  const int n = (kc < PROP) ? kc : PROP;
  for (int p = tid; p < PROP; p += blockDim.x) {
    float v0 = 0.f, v1 = 0.f, v2 = 0.f, v3 = 0.f;
    if (p < n) {
      const float* s = dec + ((size_t)b * KSEL + keep[p]) * 4;
      v0 = s[0]; v1 = s[1]; v2 = s[2]; v3 = s[3];
    }
    float* o = out + ((size_t)b * PROP + p) * 4;
    o[0] = v0; o[1] = v1; o[2] = v2; o[3] = v3;
  }
}

// ---------------------------------------------------------------------------
extern "C" void kernel_launch(void* const* d_in, const int* in_sizes, int n_in,
                              void* d_out, int out_size, void* d_ws, size_t ws_size,
                              hipStream_t stream) {
  (void)in_sizes; (void)n_in; (void)out_size; (void)ws_size;
  const float* classes = (const float*)d_in[0];   // (B, N, 2) f32
  const float* bboxes  = (const float*)d_in[1];   // (B, N, 4) f32
  const float* anchors = (const float*)d_in[2];   // (B, N, 4) f32
  float* out = (float*)d_out;                     // (B, 1000, 4) f32

  // workspace layout (256B aligned):
  //   state : B*4 u32                         @ 0
  //   hist  : B*256 u32                       @ 256
  //   cand  : B*8192 u64                      @ 4608
  //   dec   : B*6000*4 f32                    @ 266752
  //   mask  : B*6000*188 u32 (~18.05 MB)      @ 650752
  char* ws = (char*)d_ws;
  unsigned*           state = (unsigned*)(ws + 0);
  unsigned*           hist  = (unsigned*)(ws + 256);
  unsigned long long* cand  = (unsigned long long*)(ws + 4608);
  float*              dec   = (float*)(ws + 266752);
  unsigned*           mask  = (unsigned*)(ws + 650752);

  init_kernel<<<1, 1024, 0, stream>>>(state, hist);
  for (int pass = 0; pass < 4; ++pass) {
    hist_kernel<<<dim3(128, BATCH), 256, 0, stream>>>(classes, hist, state, pass);
    scan_kernel<<<BATCH, 256, 0, stream>>>(hist, state, pass);
  }
  compact_kernel<<<dim3(128, BATCH), 256, 0, stream>>>(classes, state, cand);
  sort_decode_kernel<<<BATCH, 1024, CANDN * sizeof(unsigned long long), stream>>>(
      cand, anchors, bboxes, state, dec);
  mask_kernel<<<dim3(MWU, KSEL / 16, BATCH), 32, 0, stream>>>(dec, mask);
  nms_out_kernel<<<BATCH, 256, 0, stream>>>(dec, mask, out);
}